// PfAAMLayer_35622458753953
// MI455X (gfx1250) — compile-verified
//
#include <hip/hip_runtime.h>
#include <cstdint>
#include <cstddef>

// Problem geometry (fixed by the reference): x[8,32,32,64,64] f32
#define BDIM 8
#define CDIM 32
#define TDIM 32
#define HDIM 64
#define WDIM 64
#define S    (TDIM * HDIM * WDIM)   // 131072 spatial elems per (b,c)
#define NBC  (BDIM * CDIM)          // 256 (b,c) pairs
#define POS  64                     // spatial positions per tile in pass B
#define E_LAMBDA 1e-4f

// ---------------------------------------------------------------------------
// Pass A: per-(b,c) sum / sum-of-squares -> mu and 1/denom into workspace.
// One block per (b,c); fixed 128-trip loop of float4 loads (no exec-mask
// loop-exit overhead), fma accumulation, wave32 shuffle reduction.
// RT temporal hint leaves x resident in the 192MB L2 for pass B.
// ---------------------------------------------------------------------------
__global__ __launch_bounds__(256) void pass_stats(const float* __restrict__ x,
                                                  float* __restrict__ ws) {
  const int bc  = blockIdx.x;
  const int tid = threadIdx.x;
  const float4* xp = (const float4*)(x + (size_t)bc * S);

  float s = 0.f, q = 0.f;
  #pragma unroll 8
  for (int it = 0; it < (S / 4) / 256; ++it) {     // 128 uniform iterations
    float4 v = xp[it * 256 + tid];
    s += v.x + v.y + v.z + v.w;
    q = fmaf(v.x, v.x, q);
    q = fmaf(v.y, v.y, q);
    q = fmaf(v.z, v.z, q);
    q = fmaf(v.w, v.w, q);
  }

  // wave32 butterfly, then combine the 8 waves through LDS
  #pragma unroll
  for (int off = 16; off > 0; off >>= 1) {
    s += __shfl_down(s, off, 32);
    q += __shfl_down(q, off, 32);
  }
  __shared__ float rs[8];
  __shared__ float rq[8];
  if ((tid & 31) == 0) { rs[tid >> 5] = s; rq[tid >> 5] = q; }
  __syncthreads();

  if (tid == 0) {
    float sum = 0.f, sumsq = 0.f;
    #pragma unroll
    for (int wv = 0; wv < 8; ++wv) { sum += rs[wv]; sumsq += rq[wv]; }
    const float mu    = sum * (1.0f / (float)S);
    const float sd2   = fmaf(-(float)S * mu, mu, sumsq);      // sum (x-mu)^2
    const float nrm   = 1.0f / (float)(HDIM * WDIM - 1);      // n = h*w-1 = 4095
    const float denom = 4.0f * fmaf(sd2, nrm, E_LAMBDA);
    ws[bc]       = mu;
    ws[NBC + bc] = 1.0f / denom;
  }
}

// ---------------------------------------------------------------------------
// Pass B: each block owns POS=64 spatial positions; async-loads the full
// [256 x 64] f32 cross-section (64KB) into LDS via the CDNA5 async
// global->LDS path (ASYNCcnt), computes z (mean over c) and s (mean over b)
// in LDS, then writes the fused gated output with non-temporal stores so the
// output stream does not evict x from L2.
// ---------------------------------------------------------------------------
__global__ __launch_bounds__(256) void pass_apply(const float* __restrict__ x,
                                                  const float* __restrict__ ws,
                                                  float* __restrict__ out) {
  __shared__ float tile[NBC * POS];   // 64 KB: tile[bc][pos]
  __shared__ float zsh[BDIM * POS];   //  2 KB: mean over c, z[b][pos]
  __shared__ float ssh[CDIM * POS];   //  8 KB: mean over b, s[c][pos]
  __shared__ float mush[NBC];
  __shared__ float invsh[NBC];

  const int tid  = threadIdx.x;
  const int pos0 = blockIdx.x * POS;

  // stage per-(b,c) stats into LDS
  mush[tid]  = ws[tid];
  invsh[tid] = ws[NBC + tid];

  // --- async global->LDS tile load: 4096 chunks of 16B, 16 per thread -----
  // chunk k: row = k/16, float offset in row = (k%16)*4; consecutive lanes
  // take consecutive 16B global chunks (coalesced 256B runs per row).
  const uint32_t lds_base = (uint32_t)(uintptr_t)(&tile[0]);
  const uint64_t gbase    = (uint64_t)(uintptr_t)x;
  #pragma unroll
  for (int it = 0; it < 16; ++it) {
    const int k   = it * 256 + tid;
    const int row = k >> 4;
    const int cin = k & 15;
    const uint32_t voff  = ((uint32_t)row * (uint32_t)S + (uint32_t)pos0) * 4u
                         + (uint32_t)cin * 16u;               // byte offset, < 2^28
    const uint32_t laddr = lds_base + (uint32_t)k * 16u;
    asm volatile("global_load_async_to_lds_b128 %0, %1, %2"
                 :
                 : "v"(laddr), "v"(voff), "s"(gbase)
                 : "memory");
  }
  asm volatile("s_wait_asynccnt 0" ::: "memory");   // drain this wave's ASYNCcnt
  __syncthreads();                                   // and everyone else's

  // --- z[b][pos] = mean over c (512 values, 2/thread) ---------------------
  #pragma unroll
  for (int it = 0; it < 2; ++it) {
    const int idx = it * 256 + tid;
    const int b = idx >> 6, p = idx & 63;
    float a = 0.f;
    #pragma unroll
    for (int c = 0; c < CDIM; ++c) a += tile[(b * CDIM + c) * POS + p];
    zsh[idx] = a * (1.0f / (float)CDIM);
  }
  // --- s[c][pos] = mean over b (2048 values, 8/thread) --------------------
  #pragma unroll
  for (int it = 0; it < 8; ++it) {
    const int idx = it * 256 + tid;
    const int c = idx >> 6, p = idx & 63;
    float a = 0.f;
    #pragma unroll
    for (int b = 0; b < BDIM; ++b) a += tile[(b * CDIM + c) * POS + p];
    ssh[idx] = a * (1.0f / (float)BDIM);
  }
  __syncthreads();

  // --- fused gate + output, 4 rows per iteration, coalesced NT stores -----
  const int p    = tid & 63;
  const int rsub = tid >> 6;                  // 0..3
  #pragma unroll 2
  for (int rg = 0; rg < NBC / 4; ++rg) {
    const int r = rg * 4 + rsub;              // bc row
    const int b = r >> 5, c = r & 31;
    const float xv  = tile[r * POS + p];
    const float mu  = mush[r];
    const float inv = invsh[r];
    const float d   = xv - mu;
    const float y   = fmaf(d * d, inv, 0.5f);
    const float g1  = 1.0f / (1.0f + __expf(-y));
    const float yzs = y * zsh[b * POS + p] * ssh[c * POS + p];
    const float g2  = 1.0f / (1.0f + __expf(-yzs));
    __builtin_nontemporal_store(xv * g1 * g2, &out[(size_t)r * S + pos0 + p]);
  }
}

// ---------------------------------------------------------------------------
extern "C" void kernel_launch(void* const* d_in, const int* in_sizes, int n_in,
                              void* d_out, int out_size, void* d_ws, size_t ws_size,
                              hipStream_t stream) {
  const float* x = (const float*)d_in[0];
  float* out = (float*)d_out;
  float* ws  = (float*)d_ws;   // 512 floats: mu[256] then 1/denom[256]

  pass_stats<<<NBC, 256, 0, stream>>>(x, ws);
  pass_apply<<<S / POS, 256, 0, stream>>>(x, ws, out);
}